// PGNN_16200616641063
// MI455X (gfx1250) — compile-verified
//
#include <hip/hip_runtime.h>

// CDNA5 / gfx1250: wave32, WMMA (not MFMA). FP32 16x16x4 WMMA keeps the
// reference's fp32 numerics. The workload is bound by ~2.6 GB of edge
// aggregation traffic (~110us at 23.3 TB/s HBM), not by the 7.5 GFLOP of
// GEMM, so staying in fp32 matrix ops is the right accuracy/perf tradeoff.
//
// GEMM loads are branch-free: out-of-range rows/cols are *clamped* (still
// valid addresses; their products only land in store-masked D elements),
// and the K%4 tail masks values (not loads) with a 0/1 multiplier.

typedef float v2f __attribute__((ext_vector_type(2)));
typedef float v8f __attribute__((ext_vector_type(8)));

#define DIM_F0  54
#define DIM_F2  108
#define DIM_F3  216
#define DIM_OUT 200

static __device__ __forceinline__ long long gtid() {
  return (long long)blockIdx.x * blockDim.x + threadIdx.x;
}

__global__ void k_fill_f32(float* __restrict__ p, long long n, float v) {
  long long i = gtid();
  if (i < n) p[i] = v;
}

// deg[dst] += 1 per edge (deg pre-initialized to 1.0 for the self loop)
__global__ void k_deg(const int* __restrict__ dstI, float* __restrict__ deg, int E) {
  long long e = gtid();
  if (e < E) atomicAdd(&deg[dstI[e]], 1.0f);
}

__global__ void k_rsqrt_inplace(float* __restrict__ d, int n) {
  long long i = gtid();
  if (i < n) d[i] = rsqrtf(d[i]);
}

// Y[N,M] = X[N,K] @ W[K,M] (+ optional bias), via V_WMMA_F32_16X16X4_F32.
// One wave per 16x16 output tile. K, M compile-time for static unrolling.
template <int K, int M>
__global__ void k_gemm_wmma_f32(const float* __restrict__ X, const float* __restrict__ W,
                                const float* __restrict__ bias, float* __restrict__ Y,
                                int N, int tilesC, int totalTiles) {
  static_assert((K % 2) == 0, "K must be even");
  int wave = threadIdx.x >> 5;
  int lane = threadIdx.x & 31;
  int tile = blockIdx.x * (blockDim.x >> 5) + wave;
  if (tile >= totalTiles) return;          // wave-uniform: EXEC all-1s at every WMMA
  int rowTile = tile / tilesC;
  int colTile = tile - rowTile * tilesC;
  int half = lane >> 4;                    // 0: lanes 0-15, 1: lanes 16-31
  int lr   = lane & 15;
  int row  = rowTile * 16 + lr;            // A: lane holds matrix row m
  int col  = colTile * 16 + lr;            // B/D: lane holds matrix col n
  int rowc = row < N ? row : N - 1;        // clamped: always-valid load address
  int colc = col < M ? col : M - 1;
  const float* __restrict__ xrow = X + (size_t)rowc * K;

  v8f acc = {};
  constexpr int KA = (K / 4) * 4;          // iterations with all lanes fully in-range
#pragma unroll
  for (int k0 = 0; k0 < KA; k0 += 4) {
    int ka = k0 + 2 * half;                // ISA layout: VGPR0->k0+2h, VGPR1->k0+2h+1
    v2f a, b;
    a.x = xrow[ka];
    a.y = xrow[ka + 1];
    b.x = W[(size_t)ka * M + colc];
    b.y = W[(size_t)(ka + 1) * M + colc];
    // (neg_a, A, neg_b, B, c_mod, C, reuse_a, reuse_b)
    acc = __builtin_amdgcn_wmma_f32_16x16x4_f32(false, a, false, b, (short)0, acc,
                                                false, false);
  }
  if constexpr ((K % 4) != 0) {            // K%4==2 tail: mask values, not loads
    int ka = KA + 2 * half;                // {K-2,K-1} for half 0; {K,K+1} for half 1
    float s = (ka < K) ? 1.0f : 0.0f;
    int kc = (ka < K) ? ka : K - 2;        // clamped: valid address
    v2f a, b;
    a.x = xrow[kc] * s;
    a.y = xrow[kc + 1] * s;
    b.x = W[(size_t)kc * M + colc] * s;
    b.y = W[(size_t)(kc + 1) * M + colc] * s;
    acc = __builtin_amdgcn_wmma_f32_16x16x4_f32(false, a, false, b, (short)0, acc,
                                                false, false);
  }

  if (col < M) {
    float bb = bias ? bias[col] : 0.0f;
    int mbase = rowTile * 16 + 8 * half;   // D: VGPR r -> m = r + 8*half
#pragma unroll
    for (int r = 0; r < 8; ++r) {
      int m = mbase + r;
      if (m < N) Y[(size_t)m * M + col] = acc[r] + bb;
    }
  }
}

// out[i,f] = h[i,f] * dinv[i]^2   (self-loop contribution initializes accumulator)
template <int F>
__global__ void k_selfinit(const float* __restrict__ h, const float* __restrict__ dinv,
                           float* __restrict__ out, long long total) {
  long long t = gtid();
  if (t >= total) return;
  int i = (int)(t / F);                    // constant divisor -> mul/shift
  float di = dinv[i];
  out[t] = h[t] * di * di;
}

// out[dst,f..f+1] += h[src,f..f+1] * dinv[src]*dinv[dst]; float2-vectorized gather
template <int F>
__global__ void k_edge_scatter(const float* __restrict__ h, const int* __restrict__ srcI,
                               const int* __restrict__ dstI, const float* __restrict__ dinv,
                               float* __restrict__ out, long long total) {
  static_assert((F % 2) == 0, "F must be even");
  constexpr int H = F / 2;
  long long t = gtid();
  if (t >= total) return;
  int e = (int)(t / H);                    // constant divisor -> mul/shift
  int f = 2 * (int)(t - (long long)e * H);
  int s = srcI[e];
  int d = dstI[e];
  float nm = dinv[s] * dinv[d];
  const float2 hv = *(const float2*)(h + (size_t)s * F + f);
  atomicAdd(&out[(size_t)d * F + f],     hv.x * nm);
  atomicAdd(&out[(size_t)d * F + f + 1], hv.y * nm);
}

template <int F>
__global__ void k_bias_relu(float* __restrict__ h, const float* __restrict__ b,
                            long long total) {
  long long t = gtid();
  if (t >= total) return;
  int f = (int)(t % F);
  float v = h[t] + b[f];
  h[t] = v > 0.0f ? v : 0.0f;
}

// batch is sorted: record the first node index of each (non-empty) graph
__global__ void k_starts(const int* __restrict__ batch, int* __restrict__ starts, int N) {
  long long i = gtid();
  if (i >= N) return;
  if (i == 0 || batch[i] != batch[i - 1]) starts[batch[i]] = (int)i;
}

// dense[g, pos, f] = h[i, f]  with pos = i - starts[g]; dropped if pos >= max_num
__global__ void k_dense_scatter(const float* __restrict__ h, const int* __restrict__ batch,
                                const int* __restrict__ starts, const int* __restrict__ maxnum,
                                float* __restrict__ out, long long total) {
  long long t = gtid();
  if (t >= total) return;
  long long i = t / DIM_OUT;
  int f = (int)(t - i * DIM_OUT);
  int mn = maxnum[0];
  int g = batch[i];
  int p = (int)i - starts[g];
  if (p < mn) out[((size_t)g * mn + p) * DIM_OUT + f] = h[t];
}

static inline unsigned cdiv(long long a, long long b) { return (unsigned)((a + b - 1) / b); }

extern "C" void kernel_launch(void* const* d_in, const int* in_sizes, int n_in,
                              void* d_out, int out_size, void* d_ws, size_t ws_size,
                              hipStream_t stream) {
  const float* x      = (const float*)d_in[0];
  const int*   ei     = (const int*)d_in[1];
  const int*   batch  = (const int*)d_in[2];
  const int*   maxnum = (const int*)d_in[3];
  const float* W1 = (const float*)d_in[4];  const float* b1 = (const float*)d_in[5];
  const float* W2 = (const float*)d_in[6];  const float* b2 = (const float*)d_in[7];
  const float* W3 = (const float*)d_in[8];  const float* b3 = (const float*)d_in[9];
  const float* Wf = (const float*)d_in[10]; const float* bf = (const float*)d_in[11];
  float* outp = (float*)d_out;

  const int N = in_sizes[2];          // 50000 nodes
  const int E = in_sizes[1] / 2;      // 800000 edges
  const int* srcI = ei;
  const int* dstI = ei + E;

  // workspace layout (256B-aligned chunks)
  float* ws   = (float*)d_ws;
  size_t off0 = ((size_t)N + 63) & ~(size_t)63;
  size_t hbuf = (((size_t)N * DIM_F3) + 63) & ~(size_t)63;
  float* dinv = ws;
  float* bufA = ws + off0;
  float* bufB = bufA + hbuf;
  int*   starts = (int*)(bufB + hbuf);

  const int TPB = 256;                // 8 wave32 waves per block

  // ---- symmetric GCN norm: deg (with self loop) -> dinv ----
  k_fill_f32<<<cdiv(N, TPB), TPB, 0, stream>>>(dinv, N, 1.0f);
  k_deg<<<cdiv(E, TPB), TPB, 0, stream>>>(dstI, dinv, E);
  k_rsqrt_inplace<<<cdiv(N, TPB), TPB, 0, stream>>>(dinv, N);

#define GEMM(K_, M_, Xp, Wp, bp, Yp)                                              \
  do {                                                                            \
    int tilesR = (N + 15) / 16;                                                   \
    int tilesC = ((M_) + 15) / 16;                                                \
    int total  = tilesR * tilesC;                                                 \
    k_gemm_wmma_f32<K_, M_><<<cdiv(total, TPB / 32), TPB, 0, stream>>>(           \
        (Xp), (Wp), (bp), (Yp), N, tilesC, total);                                \
  } while (0)

#define AGGREGATE(F_, hin, hout, bp)                                              \
  do {                                                                            \
    long long tn = (long long)N * (F_);                                           \
    long long te = (long long)E * ((F_) / 2);                                     \
    k_selfinit<F_><<<cdiv(tn, TPB), TPB, 0, stream>>>((hin), dinv, (hout), tn);   \
    k_edge_scatter<F_><<<cdiv(te, TPB), TPB, 0, stream>>>((hin), srcI, dstI,      \
                                                          dinv, (hout), te);      \
    k_bias_relu<F_><<<cdiv(tn, TPB), TPB, 0, stream>>>((hout), (bp), tn);         \
  } while (0)

  // layer 1: x @ W1 -> aggregate -> relu(+b1)
  GEMM(DIM_F0, DIM_F0, x, W1, (const float*)nullptr, bufA);
  AGGREGATE(DIM_F0, bufA, bufB, b1);
  // layer 2
  GEMM(DIM_F0, DIM_F2, bufB, W2, (const float*)nullptr, bufA);
  AGGREGATE(DIM_F2, bufA, bufB, b2);
  // layer 3
  GEMM(DIM_F2, DIM_F3, bufB, W3, (const float*)nullptr, bufA);
  AGGREGATE(DIM_F3, bufA, bufB, b3);
  // final linear (+bf fused into the WMMA store)
  GEMM(DIM_F3, DIM_OUT, bufB, Wf, bf, bufA);

  // ---- to_dense_batch ----
  k_fill_f32<<<cdiv(out_size, TPB), TPB, 0, stream>>>(outp, out_size, 0.0f);
  k_starts<<<cdiv(N, TPB), TPB, 0, stream>>>(batch, starts, N);
  long long td = (long long)N * DIM_OUT;
  k_dense_scatter<<<cdiv(td, TPB), TPB, 0, stream>>>(bufA, batch, starts, maxnum, outp, td);
}